// PointNetSetAbstraction_82841329205662
// MI455X (gfx1250) — compile-verified
//
#include <hip/hip_runtime.h>
#include <stdint.h>

static constexpr int kB  = 8;
static constexpr int kN  = 16384;
static constexpr int kD  = 64;
static constexpr int kS  = 1024;              // NPOINT
static constexpr int kNS = 32;                // NSAMPLE
static constexpr int kM  = kB * kS * kNS;     // 262144 rows of the MLP GEMMs
static constexpr float kR2  = 0.01f;          // RADIUS^2
static constexpr float kEPS = 1e-5f;

typedef __attribute__((ext_vector_type(2))) float        v2f;
typedef __attribute__((ext_vector_type(8))) float        v8f;
typedef __attribute__((ext_vector_type(4))) unsigned int v4u;
typedef __attribute__((ext_vector_type(4))) int          v4i;
typedef __attribute__((ext_vector_type(8))) int          v8i;

// ---------------------------------------------------------------- zero init
__global__ void zero_kernel(float* a, int na, float* b, int nb) {
  int i = blockIdx.x * blockDim.x + threadIdx.x;
  if (i < na) a[i] = 0.f;
  if (i < nb) b[i] = 0.f;
}

// ---------------------------------------------------------------- TDM helper
// Build a Tensor DMA Descriptor (D#) for a 2D tile load (global -> LDS) per
// CDNA5 ISA and issue TENSOR_LOAD_TO_LDS.
//  - data_size = 4 bytes, count=1, type=2 ("image"), no padding/iterate.
//  - tensor_dim0 = width_elems, tile_dim0 = tile_w: tile_w > width gives
//    hardware zero-fill of the out-of-bounds columns (free K padding).
//  - rows: tensor_dim1 = tile_dim1 = tile_h; row stride = stride_elems.
// Descriptors must live in SGPRs -> force every word through readfirstlane
// (all call sites pass wave-uniform values).
__device__ __forceinline__ int rfl(unsigned int x) {
  return __builtin_amdgcn_readfirstlane((int)x);
}

__device__ __forceinline__ void tdm_load_2d(uint32_t lds_addr, const float* gptr,
                                            int width_elems, int tile_w,
                                            int tile_h, int stride_elems) {
  const uint64_t ga = (uint64_t)(uintptr_t)gptr;
  v4u g0;
  g0.x = (unsigned)rfl(1u);                                   // count=1 (valid, user mode)
  g0.y = (unsigned)rfl(lds_addr);                             // lds_addr[31:0]
  g0.z = (unsigned)rfl((uint32_t)(ga & 0xFFFFFFFFu));         // global_addr[31:0]
  g0.w = (unsigned)rfl(((uint32_t)(ga >> 32) & 0x01FFFFFFu) | (2u << 30)); // ga[56:32]|type=2

  v8i g1;
  g1[0] = rfl(2u << 16);                                      // data_size=2 (4B), mask=0
  g1[1] = rfl(((unsigned)width_elems & 0xFFFFu) << 16);       // tensor_dim0[15:0] @63:48
  g1[2] = rfl((((unsigned)width_elems >> 16) & 0xFFFFu) |     // tensor_dim0[31:16]
              (((unsigned)tile_h & 0xFFFFu) << 16));          // tensor_dim1[15:0]
  g1[3] = rfl((((unsigned)tile_h >> 16) & 0xFFFFu) |          // tensor_dim1[31:16]
              (((unsigned)tile_w & 0xFFFFu) << 16));          // tile_dim0
  g1[4] = rfl((unsigned)tile_h & 0xFFFFu);                    // tile_dim1, tile_dim2=0
  g1[5] = rfl((unsigned)stride_elems);                        // tensor_dim0_stride[31:0]
  g1[6] = rfl(0u);                                            // stride hi | dim1_stride lo
  g1[7] = rfl(0u);                                            // dim1_stride hi (unused, 2D)

  const v4i gz4 = {0, 0, 0, 0};
#if __clang_major__ >= 23
  const v8i gz8 = {0, 0, 0, 0, 0, 0, 0, 0};
  __builtin_amdgcn_tensor_load_to_lds(g0, g1, gz4, gz4, gz8, 0);
#else
  __builtin_amdgcn_tensor_load_to_lds(g0, g1, gz4, gz4, 0);
#endif
}

// ---------------------------------------------------------------- FPS
// One workgroup (1024 threads = 32 wave32) per batch. Each thread owns 16
// points; running min-distances stay in registers. Two-level argmax
// (shuffle within wave, LDS across waves) with lowest-index tie-break to
// match jnp.argmax. Sampled indices kept in LDS history.
__global__ void __launch_bounds__(1024) fps_kernel(const float* __restrict__ xyz,
                                                   float* __restrict__ new_xyz) {
  const int b   = blockIdx.x;
  const int tid = threadIdx.x;
  const float* xb = xyz + (size_t)b * 3 * kN;

  float px[16], py[16], pz[16], pd[16];
#pragma unroll
  for (int j = 0; j < 16; ++j) {
    int i = tid + j * 1024;
    px[j] = xb[i];
    py[j] = xb[kN + i];
    pz[j] = xb[2 * kN + i];
    pd[j] = 1e10f;
  }

  __shared__ float s_val[32];
  __shared__ int   s_idx[32];
  __shared__ float s_c[3];
  __shared__ int   s_cur;
  __shared__ int   s_hist[kS];

  int cur = 0;
  for (int it = 0; it < kS; ++it) {
    if (tid == 0) s_hist[it] = cur;
    if (tid == (cur & 1023)) {           // owner broadcasts centroid coords
      int j = cur >> 10;
      s_c[0] = px[j]; s_c[1] = py[j]; s_c[2] = pz[j];
    }
    __syncthreads();
    const float cx = s_c[0], cy = s_c[1], cz = s_c[2];

    float bv = -1.f;
    int   bi = 0;
#pragma unroll
    for (int j = 0; j < 16; ++j) {
      float dx = px[j] - cx, dy = py[j] - cy, dz = pz[j] - cz;
      float d  = dx * dx + dy * dy + dz * dz;
      float nd = fminf(pd[j], d);
      pd[j] = nd;
      if (nd > bv) { bv = nd; bi = tid + j * 1024; }  // j ascending => lowest idx on tie
    }
    // wave32 argmax, lowest index wins ties
#pragma unroll
    for (int off = 16; off > 0; off >>= 1) {
      float ov = __shfl_down(bv, off);
      int   oi = __shfl_down(bi, off);
      if (ov > bv || (ov == bv && oi < bi)) { bv = ov; bi = oi; }
    }
    __syncthreads();
    if ((tid & 31) == 0) { s_val[tid >> 5] = bv; s_idx[tid >> 5] = bi; }
    __syncthreads();
    if (tid < 32) {                      // wave 0 reduces the 32 wave results
      bv = s_val[tid]; bi = s_idx[tid];
#pragma unroll
      for (int off = 16; off > 0; off >>= 1) {
        float ov = __shfl_down(bv, off);
        int   oi = __shfl_down(bi, off);
        if (ov > bv || (ov == bv && oi < bi)) { bv = ov; bi = oi; }
      }
      if (tid == 0) s_cur = bi;
    }
    __syncthreads();
    cur = s_cur;
  }
  __syncthreads();
  const int id = s_hist[tid];            // tid == sample index s (0..1023)
  new_xyz[((size_t)b * 3 + 0) * kS + tid] = xb[id];
  new_xyz[((size_t)b * 3 + 1) * kS + tid] = xb[kN + id];
  new_xyz[((size_t)b * 3 + 2) * kS + tid] = xb[2 * kN + id];
}

// ---------------------------------------------------------------- ball query
// One wave32 per centroid: scan points in index order, collect first 32
// in-radius indices (== sort-then-truncate of the reference), pad with the
// first hit.
__global__ void ball_kernel(const float* __restrict__ xyz,
                            const float* __restrict__ new_xyz,
                            int* __restrict__ ball_idx) {
  const int gw   = (blockIdx.x * blockDim.x + threadIdx.x) >> 5;  // centroid id
  const int lane = threadIdx.x & 31;
  const int b = gw >> 10;
  const int s = gw & (kS - 1);
  const float* xb = xyz + (size_t)b * 3 * kN;
  const float cx = new_xyz[((size_t)b * 3 + 0) * kS + s];
  const float cy = new_xyz[((size_t)b * 3 + 1) * kS + s];
  const float cz = new_xyz[((size_t)b * 3 + 2) * kS + s];
  int* out = ball_idx + (size_t)gw * kNS;

  int cnt = 0;
  int first = 0;
  for (int i0 = 0; i0 < kN && cnt < kNS; i0 += 32) {
    const int i = i0 + lane;
    const float dx = xb[i] - cx, dy = xb[kN + i] - cy, dz = xb[2 * kN + i] - cz;
    const bool in = (dx * dx + dy * dy + dz * dz) <= kR2;
    const unsigned m = (unsigned)__ballot((int)in);
    if (cnt == 0 && m) first = i0 + __ffs((int)m) - 1;
    const int pos = cnt + __popc(m & ((1u << lane) - 1u));
    if (in && pos < kNS) out[pos] = i;
    cnt += __popc(m);
  }
  if (cnt > kNS) cnt = kNS;
  for (int p = cnt + lane; p < kNS; p += 32) out[p] = first;  // pad with first
}

// ---------------------------------------------------------------- gather
// feat row m = ((b*S+s)*32+n): [0..2]=xyz[idx]-new_xyz, [3..66]=points[idx],
// [67]=0 pad so K is a multiple of 4 for WMMA. One wave per row.
__global__ void gather_kernel(const float* __restrict__ xyz,
                              const float* __restrict__ points,
                              const float* __restrict__ new_xyz,
                              const int* __restrict__ ball_idx,
                              float* __restrict__ feat) {
  const int m    = (blockIdx.x * blockDim.x + threadIdx.x) >> 5;
  const int lane = threadIdx.x & 31;
  const int group = m >> 5;            // b*S + s
  const int s = group & (kS - 1);
  const int b = group >> 10;
  const int gi = ball_idx[m];
  float* row = feat + (size_t)m * 68;
  const float* xb = xyz + (size_t)b * 3 * kN;
  const float* pb = points + (size_t)b * kD * kN;
  if (lane < 3) row[lane] = xb[(size_t)lane * kN + gi] -
                            new_xyz[((size_t)b * 3 + lane) * kS + s];
  for (int c = lane; c < kD; c += 32) row[3 + c] = pb[(size_t)c * kN + gi];
  if (lane == 0) row[67] = 0.f;
}

// ---------------------------------------------------------------- WMMA core
// One wave computes a 16x16 f32 tile of y = A*W^T + bias using
// V_WMMA_F32_16X16X4_F32, reading A/B fragments from LDS tiles staged by the
// Tensor Data Mover.
// A fragment (16x4): lanes 0-15/16-31 both hold row M=lane&15,
//   VGPR0/1 = K = k0+2*half +{0,1}.   B fragment mirrors with N=lane&15.
template <int KPAD>
__device__ __forceinline__ v8f wmma_tile_lds(const float* __restrict__ sA,
                                             const float* __restrict__ sB,
                                             float bv, int lane) {
  const int col  = lane & 15;
  const int half = lane >> 4;
  v8f c;
#pragma unroll
  for (int r = 0; r < 8; ++r) c[r] = bv;
  const float* arow = sA + col * KPAD;
  const float* brow = sB + col * KPAD;
  for (int k0 = 0; k0 < KPAD; k0 += 4) {
    const int ka = k0 + 2 * half;
    v2f a = *(const v2f*)(arow + ka);
    v2f b = *(const v2f*)(brow + ka);
    c = __builtin_amdgcn_wmma_f32_16x16x4_f32(false, a, false, b, (short)0, c,
                                              false, false);
  }
  return c;
}

// GEMM (+ optional store) + per-channel sum/sumsq accumulation for BN stats.
// TDM stages: wave0 DMAs the 16xKREAL weight tile (zero-padded to KPAD by
// tensor OOB reads); every wave DMAs its own contiguous 16xKPAD A tile.
template <int KREAL, int KPAD, int COUT, bool STORE>
__global__ void gemm_stats_kernel(const float* __restrict__ A,
                                  const float* __restrict__ W,
                                  const float* __restrict__ bias,
                                  float* __restrict__ out,
                                  float* __restrict__ stats) {
  __shared__ __align__(16) float sB[16 * KPAD];
  __shared__ __align__(16) float sA[8][16 * KPAD];
  __shared__ float ls[16];
  __shared__ float lq[16];
  const int lane  = threadIdx.x & 31;
  const int wave  = threadIdx.x >> 5;
  const int tileN = blockIdx.y;
  const int tileM = blockIdx.x * 8 + wave;

  if (wave == 0) {
    tdm_load_2d((uint32_t)(uintptr_t)&sB[0], W + (size_t)tileN * 16 * KREAL,
                KREAL, KPAD, 16, KREAL);
  }
  tdm_load_2d((uint32_t)(uintptr_t)&sA[wave][0], A + (size_t)tileM * 16 * KPAD,
              KPAD, KPAD, 16, KPAD);
  if (threadIdx.x < 16) { ls[threadIdx.x] = 0.f; lq[threadIdx.x] = 0.f; }
  __builtin_amdgcn_s_wait_tensorcnt(0);
  __syncthreads();                       // wave0's sB visible to all waves

  const int col  = lane & 15;
  const int half = lane >> 4;
  v8f c = wmma_tile_lds<KPAD>(&sA[wave][0], &sB[0], bias[tileN * 16 + col], lane);

  if constexpr (STORE) {
    float* orow = out + (size_t)tileM * 16 * COUT + tileN * 16 + col;
#pragma unroll
    for (int r = 0; r < 8; ++r) orow[(size_t)(r + 8 * half) * COUT] = c[r];
  }
  float s = 0.f, q = 0.f;
#pragma unroll
  for (int r = 0; r < 8; ++r) { s += c[r]; q += c[r] * c[r]; }
  s += __shfl_xor(s, 16);   // lane l and l^16 hold the same channel column
  q += __shfl_xor(q, 16);
  if (half == 0) { atomicAdd(&ls[col], s); atomicAdd(&lq[col], q); }
  __syncthreads();
  if (threadIdx.x < 16) {
    atomicAdd(&stats[tileN * 16 + threadIdx.x], ls[threadIdx.x]);
    atomicAdd(&stats[COUT + tileN * 16 + threadIdx.x], lq[threadIdx.x]);
  }
}

// In-place BN + ReLU using finished global stats.
template <int COUT>
__global__ void bn_relu_kernel(float* __restrict__ act,
                               const float* __restrict__ stats,
                               const float* __restrict__ g,
                               const float* __restrict__ be) {
  const size_t e = (size_t)blockIdx.x * blockDim.x + threadIdx.x;
  const int ch = (int)(e & (COUT - 1));
  const float inv  = 1.f / (float)kM;
  const float mean = stats[ch] * inv;
  const float var  = stats[COUT + ch] * inv - mean * mean;
  const float sc   = g[ch] * rsqrtf(var + kEPS);
  const float sh   = be[ch] - mean * sc;
  const float v    = act[e] * sc + sh;
  act[e] = v > 0.f ? v : 0.f;
}

// Layer-3 pass 2: recompute WMMA tile, BN+ReLU in registers, fold max over
// the tile's 16 rows (all one (b,s) group) and atomic-max (bit-pattern trick,
// valid for the non-negative post-ReLU values) into new_points (B,128,S).
template <int KREAL, int KPAD, int COUT>
__global__ void gemm_bn_max_kernel(const float* __restrict__ A,
                                   const float* __restrict__ W,
                                   const float* __restrict__ bias,
                                   const float* __restrict__ stats,
                                   const float* __restrict__ g,
                                   const float* __restrict__ be,
                                   unsigned int* __restrict__ outp) {
  __shared__ __align__(16) float sB[16 * KPAD];
  __shared__ __align__(16) float sA[8][16 * KPAD];
  const int lane  = threadIdx.x & 31;
  const int wave  = threadIdx.x >> 5;
  const int tileN = blockIdx.y;
  const int tileM = blockIdx.x * 8 + wave;

  if (wave == 0) {
    tdm_load_2d((uint32_t)(uintptr_t)&sB[0], W + (size_t)tileN * 16 * KREAL,
                KREAL, KPAD, 16, KREAL);
  }
  tdm_load_2d((uint32_t)(uintptr_t)&sA[wave][0], A + (size_t)tileM * 16 * KPAD,
              KPAD, KPAD, 16, KPAD);
  __builtin_amdgcn_s_wait_tensorcnt(0);
  __syncthreads();

  const int col = lane & 15;
  v8f c = wmma_tile_lds<KPAD>(&sA[wave][0], &sB[0], bias[tileN * 16 + col], lane);

  const int ch  = tileN * 16 + col;
  const float inv  = 1.f / (float)kM;
  const float mean = stats[ch] * inv;
  const float var  = stats[COUT + ch] * inv - mean * mean;
  const float sc   = g[ch] * rsqrtf(var + kEPS);
  const float sh   = be[ch] - mean * sc;

  float m = 0.f;
#pragma unroll
  for (int r = 0; r < 8; ++r) {
    float v = c[r] * sc + sh;
    v = v > 0.f ? v : 0.f;
    m = fmaxf(m, v);
  }
  m = fmaxf(m, __shfl_xor(m, 16));
  if (lane < 16) {
    const int group = tileM >> 1;        // (b*S + s): 16-row tile is half a group
    const int b = group >> 10;
    const int s = group & (kS - 1);
    atomicMax(&outp[((size_t)b * COUT + ch) * kS + s], __float_as_uint(m));
  }
}

// ---------------------------------------------------------------- launcher
extern "C" void kernel_launch(void* const* d_in, const int* in_sizes, int n_in,
                              void* d_out, int out_size, void* d_ws, size_t ws_size,
                              hipStream_t stream) {
  (void)in_sizes; (void)n_in; (void)out_size; (void)ws_size;
  const float* xyz    = (const float*)d_in[0];
  const float* points = (const float*)d_in[1];
  const float* w1  = (const float*)d_in[2];
  const float* b1  = (const float*)d_in[3];
  const float* g1  = (const float*)d_in[4];
  const float* be1 = (const float*)d_in[5];
  const float* w2  = (const float*)d_in[6];
  const float* b2  = (const float*)d_in[7];
  const float* g2  = (const float*)d_in[8];
  const float* be2 = (const float*)d_in[9];
  const float* w3  = (const float*)d_in[10];
  const float* b3  = (const float*)d_in[11];
  const float* g3  = (const float*)d_in[12];
  const float* be3 = (const float*)d_in[13];

  float* out        = (float*)d_out;
  float* new_xyz    = out;                         // (B,3,S)
  float* new_points = out + (size_t)kB * 3 * kS;   // (B,128,S)

  // workspace carve-up (all 8-byte aligned)
  int*   ball_idx = (int*)d_ws;                    // kM ints
  float* stats    = (float*)(ball_idx + kM);       // 512 floats: st1|st2|st3
  float* st1   = stats;        // 2*64
  float* st2   = stats + 128;  // 2*64
  float* st3   = stats + 256;  // 2*128
  float* feat0 = stats + 512;                      // kM * 68 (layer1 input)
  float* act1  = feat0 + (size_t)kM * 68;          // kM * 64
  float* act2  = feat0;                            // reuse feat0 for layer2 out

  const int np = kB * 128 * kS;                    // 1048576
  zero_kernel<<<(np + 255) / 256, 256, 0, stream>>>(stats, 512, new_points, np);

  fps_kernel<<<kB, 1024, 0, stream>>>(xyz, new_xyz);
  ball_kernel<<<(kB * kS) / 8, 256, 0, stream>>>(xyz, new_xyz, ball_idx);
  gather_kernel<<<kM / 8, 256, 0, stream>>>(xyz, points, new_xyz, ball_idx, feat0);

  const dim3 blk(256);
  // layer 1: 67 -> 64
  gemm_stats_kernel<67, 68, 64, true>
      <<<dim3(kM / 16 / 8, 4), blk, 0, stream>>>(feat0, w1, b1, act1, st1);
  bn_relu_kernel<64><<<(kM * 64) / 256, blk, 0, stream>>>(act1, st1, g1, be1);
  // layer 2: 64 -> 64
  gemm_stats_kernel<64, 64, 64, true>
      <<<dim3(kM / 16 / 8, 4), blk, 0, stream>>>(act1, w2, b2, act2, st2);
  bn_relu_kernel<64><<<(kM * 64) / 256, blk, 0, stream>>>(act2, st2, g2, be2);
  // layer 3: 64 -> 128, stats pass (no store), then fused BN+ReLU+neighbor-max
  gemm_stats_kernel<64, 64, 128, false>
      <<<dim3(kM / 16 / 8, 8), blk, 0, stream>>>(act2, w3, b3, nullptr, st3);
  gemm_bn_max_kernel<64, 64, 128>
      <<<dim3(kM / 16 / 8, 8), blk, 0, stream>>>(act2, w3, b3, st3, g3, be3,
                                                 (unsigned int*)new_points);
}